// LRU_35777077576563
// MI455X (gfx1250) — compile-verified
//
#include <hip/hip_runtime.h>

// LRU forward on gfx1250:
//   Bu = X @ Bcat            (fp32 WMMA GEMM, async-LDS double buffered)
//   h  = chunked diagonal complex scan over L (local scan / carry scan / fixup)
//   y  = h @ W2 + D*X        (same WMMA GEMM, fused epilogue)
// Sizes: H=256, N=512, L=32768.
// Workspace (floats):
//   Bu/h  : L*1024    interleaved (re,im) pairs per state      134 MB
//   Bcat  : 256*1024  (K=H rows, cols j: state j>>1, j&1=im)     1 MB
//   W2    : 1024*256  (rows n2: state n2>>1, n2&1 -> -C_im)      1 MB
//   lam   : 1024      (float2 per state)
//   carry : 128*1024  (float2 per chunk x state)
//   pref  : 128*1024

typedef __attribute__((ext_vector_type(2))) float v2f;
typedef __attribute__((ext_vector_type(8))) float v8f;

#define SEQ_L   32768
#define STATE_N 512
#define FEAT_H  256
#define CHUNK   256
#define NCHUNK  (SEQ_L / CHUNK)   // 128

// ---- CDNA5 async memory->LDS copy (ASYNCcnt-tracked, bypasses VGPRs) ------
__device__ __forceinline__ void async_copy_b128(void* lds_ptr, const float* gptr) {
    uint32_t lds = (uint32_t)(uintptr_t)lds_ptr;         // low 32 bits = LDS offset
    uint64_t ga  = (uint64_t)(uintptr_t)gptr;
    asm volatile("global_load_async_to_lds_b128 %0, %1, off"
                 :: "v"(lds), "v"(ga) : "memory");
}
__device__ __forceinline__ void wait_async0() {
    asm volatile("s_wait_asynccnt 0x0" ::: "memory");
}

// ---------------------------------------------------------------------------
// Tiled f32 WMMA GEMM: C[M,N] = A[M,K] @ B[K,N] (+ optional D[col]*X[row,col])
// Block tile 128(M) x 64(N), 256 threads = 8 waves (4x2), wave = 2x2 16x16
// tiles, K staged 16 at a time through double-buffered LDS via async copies.
// ---------------------------------------------------------------------------
__device__ __forceinline__ void issue_tile_async(
    const float* __restrict__ A, const float* __restrict__ B,
    float (*As)[20], float (*Bs)[64],
    int tid, int m_blk, int n_blk, int k0, int K, int N)
{
    // A tile: 128x16 floats = 512 b128 transfers, 2 per thread.
#pragma unroll
    for (int i = 0; i < 2; ++i) {
        int e  = tid + i * 256;          // 0..511
        int r  = e >> 2;                 // 0..127
        int c4 = (e & 3) << 2;           // 0,4,8,12
        async_copy_b128(&As[r][c4], A + (size_t)(m_blk + r) * K + (k0 + c4));
    }
    // B tile: 16x64 floats = 256 b128 transfers, 1 per thread.
    {
        int r  = tid >> 4;               // 0..15
        int c4 = (tid & 15) << 2;        // 0..60
        async_copy_b128(&Bs[r][c4], B + (size_t)(k0 + r) * N + (n_blk + c4));
    }
}

__global__ __launch_bounds__(256)
void lru_gemm_f32(const float* __restrict__ A, const float* __restrict__ B,
                  float* __restrict__ C, int M, int N, int K,
                  const float* __restrict__ Dvec, const float* __restrict__ X)
{
    // pad 20: keeps 16B alignment for b128 async stores AND kills bank
    // conflicts (20*m mod 64 distinct for m = 0..15).
    __shared__ float As[2][128][20];
    __shared__ float Bs[2][16][64];

    const int tid    = threadIdx.x;
    const int wid    = tid >> 5;
    const int lane   = tid & 31;
    const int half   = lane >> 4;        // 0: lanes 0-15, 1: lanes 16-31
    const int l16    = lane & 15;
    const int wave_m = wid >> 1;         // 0..3
    const int wave_n = wid & 1;          // 0..1
    const int m_blk  = blockIdx.x * 128;
    const int n_blk  = blockIdx.y * 64;

    const v8f vzero = {0.f, 0.f, 0.f, 0.f, 0.f, 0.f, 0.f, 0.f};
    v8f acc[2][2];
#pragma unroll
    for (int a = 0; a < 2; ++a)
#pragma unroll
        for (int b = 0; b < 2; ++b) acc[a][b] = vzero;

    const int nk = K >> 4;
    issue_tile_async(A, B, As[0], Bs[0], tid, m_blk, n_blk, 0, K, N);

    for (int ki = 0; ki < nk; ++ki) {
        wait_async0();          // tile ki landed in LDS (this wave's batch)
        __syncthreads();        // all waves landed + done computing tile ki-1
        if (ki + 1 < nk)        // overlap: fly tile ki+1 while computing ki
            issue_tile_async(A, B, As[(ki + 1) & 1], Bs[(ki + 1) & 1],
                             tid, m_blk, n_blk, (ki + 1) << 4, K, N);
        const int bsel = ki & 1;

#pragma unroll
        for (int kk = 0; kk < 16; kk += 4) {
            // A frag (ISA 7.12.2, 32-bit 16x4): lanes 0-15 K={kk,kk+1},
            // lanes 16-31 K={kk+2,kk+3}, M = lane&15.  B frag mirrors with N.
            v2f afrag[2], bfrag[2];
#pragma unroll
            for (int tm = 0; tm < 2; ++tm) {
                int m = wave_m * 32 + tm * 16 + l16;
                afrag[tm].x = As[bsel][m][kk + 2 * half];
                afrag[tm].y = As[bsel][m][kk + 2 * half + 1];
            }
#pragma unroll
            for (int tn = 0; tn < 2; ++tn) {
                int n = wave_n * 32 + tn * 16 + l16;
                bfrag[tn].x = Bs[bsel][kk + 2 * half][n];
                bfrag[tn].y = Bs[bsel][kk + 2 * half + 1][n];
            }
#pragma unroll
            for (int tm = 0; tm < 2; ++tm)
#pragma unroll
                for (int tn = 0; tn < 2; ++tn)
                    acc[tm][tn] = __builtin_amdgcn_wmma_f32_16x16x4_f32(
                        false, afrag[tm], false, bfrag[tn],
                        (short)0, acc[tm][tn], false, false);
        }
    }

    // D-matrix store: VGPR i holds row (i + 8*half), col = lane&15.
#pragma unroll
    for (int tm = 0; tm < 2; ++tm)
#pragma unroll
        for (int tn = 0; tn < 2; ++tn)
#pragma unroll
            for (int i = 0; i < 8; ++i) {
                int row = m_blk + wave_m * 32 + tm * 16 + half * 8 + i;
                int col = n_blk + wave_n * 32 + tn * 16 + l16;
                size_t off = (size_t)row * N + col;
                float v = acc[tm][tn][i];
                if (Dvec) v += Dvec[col] * X[off];
                C[off] = v;
            }
}

// ---------------------------------------------------------------------------
// Prep kernels (interleaved complex column/row order)
// ---------------------------------------------------------------------------
__global__ __launch_bounds__(256)
void lru_prep_lambda(const float* __restrict__ nu_log,
                     const float* __restrict__ theta_log,
                     float2* __restrict__ lam)
{
    int n = blockIdx.x * blockDim.x + threadIdx.x;
    if (n < STATE_N) {
        float mag = expf(-expf(nu_log[n]));
        float th  = expf(theta_log[n]);
        lam[n] = make_float2(mag * cosf(th), mag * sinf(th));
    }
}

__global__ __launch_bounds__(256)
void lru_prep_B(const float* __restrict__ B_re, const float* __restrict__ B_im,
                const float* __restrict__ gamma_log, float* __restrict__ Bcat)
{
    int e = blockIdx.x * 256 + threadIdx.x;      // < 256*1024
    int k = e >> 10;                             // 0..255 (K = H)
    int j = e & 1023;                            // col: state j>>1, j&1 = im
    int n = j >> 1;
    float g = expf(gamma_log[n]);
    Bcat[e] = ((j & 1) ? B_im : B_re)[n * FEAT_H + k] * g;
}

__global__ __launch_bounds__(256)
void lru_prep_W2(const float* __restrict__ C_re, const float* __restrict__ C_im,
                 float* __restrict__ W2)
{
    int e  = blockIdx.x * 256 + threadIdx.x;     // < 1024*256
    int n2 = e >> 8;                             // row: state n2>>1, n2&1 = im
    int i  = e & 255;                            // col 0..255 (H)
    int n  = n2 >> 1;
    W2[e] = (n2 & 1) ? -C_im[i * STATE_N + n] : C_re[i * STATE_N + n];
}

// ---------------------------------------------------------------------------
// Chunked diagonal complex scan on interleaved (re,im) buffer [L, 512 float2]
// ---------------------------------------------------------------------------
__global__ __launch_bounds__(512)
void lru_chunk_scan(float2* __restrict__ Hb, const float2* __restrict__ lam,
                    float2* __restrict__ carry)
{
    int n = threadIdx.x;             // state 0..511
    int c = blockIdx.x;              // chunk 0..127
    float2 l = lam[n];
    float hr = 0.f, hi = 0.f;
    size_t base = (size_t)c * CHUNK * STATE_N + n;
    for (int t = 0; t < CHUNK; ++t) {
        float2 bu = Hb[base];
        float nr = fmaf(l.x, hr, fmaf(-l.y, hi, bu.x));
        float ni = fmaf(l.x, hi, fmaf( l.y, hr, bu.y));
        hr = nr; hi = ni;
        Hb[base] = make_float2(hr, hi);
        base += STATE_N;
    }
    carry[(size_t)c * STATE_N + n] = make_float2(hr, hi);
}

__global__ __launch_bounds__(512)
void lru_carry_scan(const float2* __restrict__ carry,
                    const float2* __restrict__ lam, float2* __restrict__ pref)
{
    int n = threadIdx.x;
    float2 l = lam[n];
    float cr = l.x, ci = l.y;        // Lambda^CHUNK via 8 squarings (256=2^8)
#pragma unroll
    for (int s = 0; s < 8; ++s) {
        float t = cr * cr - ci * ci;
        ci = 2.f * cr * ci;
        cr = t;
    }
    float pr = 0.f, pi = 0.f;
    for (int c = 0; c < NCHUNK; ++c) {
        float2 a = carry[(size_t)c * STATE_N + n];
        float nr = fmaf(cr, pr, fmaf(-ci, pi, a.x));
        float ni = fmaf(cr, pi, fmaf( ci, pr, a.y));
        pr = nr; pi = ni;
        pref[(size_t)c * STATE_N + n] = make_float2(pr, pi);  // inclusive
    }
}

__global__ __launch_bounds__(512)
void lru_fixup(float2* __restrict__ Hb, const float2* __restrict__ lam,
               const float2* __restrict__ pref)
{
    int n = threadIdx.x;
    int c = blockIdx.x + 1;          // chunks 1..127
    float2 l = lam[n];
    float2 p = pref[(size_t)(c - 1) * STATE_N + n];
    float qr = p.x, qi = p.y;
    size_t base = (size_t)c * CHUNK * STATE_N + n;
    for (int t = 0; t < CHUNK; ++t) {
        float nr = l.x * qr - l.y * qi;      // q = Lambda^{t+1} * P_{c-1}
        float ni = l.x * qi + l.y * qr;
        qr = nr; qi = ni;
        float2 h = Hb[base];
        Hb[base] = make_float2(h.x + qr, h.y + qi);
        base += STATE_N;
    }
}

// ---------------------------------------------------------------------------
extern "C" void kernel_launch(void* const* d_in, const int* in_sizes, int n_in,
                              void* d_out, int out_size, void* d_ws, size_t ws_size,
                              hipStream_t stream) {
    const float* inputs    = (const float*)d_in[0];
    const float* nu_log    = (const float*)d_in[1];
    const float* theta_log = (const float*)d_in[2];
    const float* B_re      = (const float*)d_in[3];
    const float* B_im      = (const float*)d_in[4];
    const float* C_re      = (const float*)d_in[5];
    const float* C_im      = (const float*)d_in[6];
    const float* Dv        = (const float*)d_in[7];
    const float* gamma_log = (const float*)d_in[8];
    float* y = (float*)d_out;

    float* Bu    = (float*)d_ws;                       // [L,1024] interleaved
    float* Bcat  = Bu    + (size_t)SEQ_L * 1024;       // [256,1024]
    float* W2    = Bcat  + (size_t)FEAT_H * 1024;      // [1024,256]
    float* lam   = W2    + (size_t)1024 * FEAT_H;      // [512 float2]
    float* carry = lam   + 1024;                       // [128*512 float2]
    float* pref  = carry + (size_t)NCHUNK * 1024;      // [128*512 float2]

    lru_prep_lambda<<<2, 256, 0, stream>>>(nu_log, theta_log, (float2*)lam);
    lru_prep_B<<<1024, 256, 0, stream>>>(B_re, B_im, gamma_log, Bcat);
    lru_prep_W2<<<1024, 256, 0, stream>>>(C_re, C_im, W2);

    // GEMM1: Bu[L,1024] = inputs[L,256] @ Bcat[256,1024]
    lru_gemm_f32<<<dim3(SEQ_L / 128, 1024 / 64), 256, 0, stream>>>(
        inputs, Bcat, Bu, SEQ_L, 1024, FEAT_H, nullptr, nullptr);

    // Chunked diagonal scan (in-place on Bu -> h)
    lru_chunk_scan<<<NCHUNK, 512, 0, stream>>>((float2*)Bu, (float2*)lam,
                                               (float2*)carry);
    lru_carry_scan<<<1, 512, 0, stream>>>((const float2*)carry, (float2*)lam,
                                          (float2*)pref);
    lru_fixup<<<NCHUNK - 1, 512, 0, stream>>>((float2*)Bu, (float2*)lam,
                                              (const float2*)pref);

    // GEMM2: y[L,256] = h[L,1024] @ W2[1024,256] + D*inputs (fused epilogue)
    lru_gemm_f32<<<dim3(SEQ_L / 128, FEAT_H / 64), 256, 0, stream>>>(
        Bu, W2, y, SEQ_L, FEAT_H, 1024, Dv, inputs);
}